// myLayerRotation_84353157693800
// MI455X (gfx1250) — compile-verified
//
#include <hip/hip_runtime.h>
#include <hip/hip_bf16.h>

// CDNA5 / gfx1250: wave32, WMMA f32 16x16x4, 4 tiles (64 points) per wave-iter.
typedef __attribute__((ext_vector_type(2))) float v2f;
typedef __attribute__((ext_vector_type(8))) float v8f;

// z = x @ R^T  (z[i] = R x[i]), R from ZYX-intrinsic Euler angles in weight[3].
//
// Per 16-point tile:  D(16x16) = A(16x4) x B(4x16)
//   A[m][k] = R[m][k] (zero-padded)  -- built once per wave (SALU-scalarized)
//   B[k][n] = x[base+n][k]           -- 2 coalesced b32 loads / tile
//   D[m][n] = z[base+n][m]           -- lane n<16 holds z[base+n] in d[0..2]
// f32 fragment layout (ISA 7.12.2): lanes 0-15 hold K{0,1} in v0,v1;
// lanes 16-31 hold K{2,3}.  Store = b96 per lane (192B contiguous / tile).
//
// 4 tiles are processed per iteration so each wave keeps ~1.5KB of traffic in
// flight (8 clause-able loads sharing one base + imm offsets), covering the
// 23.3 TB/s HBM roofline even at moderate occupancy; all 4 b96 stores issue
// under one exec-mask switch.

__device__ __forceinline__ v8f wmma_tile(v2f a, v2f b) {
    v8f acc = {};
    return __builtin_amdgcn_wmma_f32_16x16x4_f32(
        false, a, false, b, (short)0, acc, false, false);
}

__global__ void rot_wmma_kernel(const float* __restrict__ x,
                                const float* __restrict__ w,
                                float* __restrict__ out,
                                int numTiles, int n) {
    // ---- Build R (wave-uniform; compiler lowers to scalar float ops) ----
    const float a = w[0], b = w[1], c = w[2];
    const float sa = sinf(a), ca = cosf(a);
    const float sb = sinf(b), cb = cosf(b);
    const float sc = sinf(c), cc = cosf(c);
    const float R00 = cc * cb, R01 = sa * sb * cc - ca * sc, R02 = ca * sb * cc + sa * sc;
    const float R10 = cb * sc, R11 = sa * sb * sc + ca * cc, R12 = ca * sb * sc - sa * cc;
    const float R20 = -sb,     R21 = sa * cb,                R22 = ca * cb;

    const unsigned lane = threadIdx.x & 31u;
    const unsigned m    = lane & 15u;          // A row / point-in-tile index
    const bool     hi   = lane >= 16u;         // upper half-wave: K in {2,3}

    // Per-lane A fragment (R row m, zero-padded for m>=3 and K=3).
    const float Rm0 = (m == 0u) ? R00 : (m == 1u) ? R10 : (m == 2u) ? R20 : 0.0f;
    const float Rm1 = (m == 0u) ? R01 : (m == 1u) ? R11 : (m == 2u) ? R21 : 0.0f;
    const float Rm2 = (m == 0u) ? R02 : (m == 1u) ? R12 : (m == 2u) ? R22 : 0.0f;
    v2f afrag;
    afrag.x = hi ? Rm2 : Rm0;                  // v0 = {K0 | K2}
    afrag.y = hi ? 0.0f : Rm1;                 // v1 = {K1 | K3=0}

    const unsigned waveId   = (blockIdx.x * blockDim.x + threadIdx.x) >> 5;
    const unsigned numWaves = (gridDim.x * blockDim.x) >> 5;

    // Per-lane element offsets (floats) within a 16-point (48-float) tile.
    const unsigned off0 = m * 3u + (hi ? 2u : 0u);  // bfrag.x source
    const unsigned off1 = m * 3u + 1u;              // bfrag.y source (lo half)

    const unsigned nt  = (unsigned)numTiles;
    const unsigned nt4 = nt & ~3u;

    // ---- Main loop: 4 consecutive tiles (64 points, 768B in / 768B out) ----
    for (unsigned t0 = waveId * 4u; t0 < nt4; t0 += numWaves * 4u) {
        const float* xt = x + (size_t)t0 * 48u;

        v2f bf[4];
#pragma unroll
        for (int i = 0; i < 4; ++i) {           // 8 loads, shared base + imm offs
            const float bx = xt[i * 48 + off0];
            const float by = xt[i * 48 + off1];
            bf[i].x = bx;
            bf[i].y = hi ? 0.0f : by;
        }

        const v8f d0 = wmma_tile(afrag, bf[0]);
        const v8f d1 = wmma_tile(afrag, bf[1]);
        const v8f d2 = wmma_tile(afrag, bf[2]);
        const v8f d3 = wmma_tile(afrag, bf[3]);

        if (!hi) {                               // one exec switch, 4 b96 stores
            float* ot = out + (size_t)t0 * 48u + m * 3u;
            ot[0]       = d0[0]; ot[1]       = d0[1]; ot[2]       = d0[2];
            ot[48 + 0]  = d1[0]; ot[48 + 1]  = d1[1]; ot[48 + 2]  = d1[2];
            ot[96 + 0]  = d2[0]; ot[96 + 1]  = d2[1]; ot[96 + 2]  = d2[2];
            ot[144 + 0] = d3[0]; ot[144 + 1] = d3[1]; ot[144 + 2] = d3[2];
        }
    }

    // ---- Leftover tiles (numTiles % 4), one tile per wave (uniform pred) ----
    if (waveId < nt - nt4) {
        const unsigned t = nt4 + waveId;
        const float* xt = x + (size_t)t * 48u;
        v2f bf;
        bf.x = xt[off0];
        const float by = xt[off1];
        bf.y = hi ? 0.0f : by;
        const v8f d = wmma_tile(afrag, bf);
        if (!hi) {
            float* ot = out + (size_t)t * 48u + m * 3u;
            ot[0] = d[0]; ot[1] = d[1]; ot[2] = d[2];
        }
    }

    // ---- Scalar tail for N % 16 (not taken for N = 8M) ----
    const int tailStart = numTiles * 16;
    const int gtid = blockIdx.x * blockDim.x + threadIdx.x;
    const int ti = tailStart + gtid;
    if (ti < n) {
        const float x0 = x[3 * ti], x1 = x[3 * ti + 1], x2 = x[3 * ti + 2];
        out[3 * ti + 0] = R00 * x0 + R01 * x1 + R02 * x2;
        out[3 * ti + 1] = R10 * x0 + R11 * x1 + R12 * x2;
        out[3 * ti + 2] = R20 * x0 + R21 * x1 + R22 * x2;
    }
}

extern "C" void kernel_launch(void* const* d_in, const int* in_sizes, int n_in,
                              void* d_out, int out_size, void* d_ws, size_t ws_size,
                              hipStream_t stream) {
    const float* x = (const float*)d_in[0];   // [N,3] float32
    const float* w = (const float*)d_in[1];   // [3,1] float32 (Euler angles)
    float* out = (float*)d_out;               // [N,3] float32

    const int n = in_sizes[0] / 3;            // number of points
    const int numTiles = n / 16;

    const int threads = 256;                  // 8 wave32 per block
    int blocks = 2048;                        // ~16K waves, ~8 unrolled iters each
    const int neededWaves = (numTiles + 3) / 4 > 0 ? (numTiles + 3) / 4 : 1;
    const int maxBlocks = (neededWaves * 32 + threads - 1) / threads;
    if (blocks > maxBlocks) blocks = maxBlocks;
    if (blocks < 1) blocks = 1;

    rot_wmma_kernel<<<blocks, threads, 0, stream>>>(x, w, out, numTiles, n);
}